// GATNetHeadsChanged3LayersLeakyReLUv3EmbeddingDim256_31628139168056
// MI455X (gfx1250) — compile-verified
//
#include <hip/hip_runtime.h>

// ---------------- problem constants ----------------
#define NN   16384       // nodes
#define EE   524288      // edges
#define DIN  256
#define HC   256         // H*C
#define TOTE (EE + NN)   // edges + self loops

typedef float v2f __attribute__((ext_vector_type(2)));
typedef float v8f __attribute__((ext_vector_type(8)));

// ---------------- workspace layout (float offsets) ----------------
#define OFF_H     ((size_t)0)                       // N*256
#define OFF_AGG   (OFF_H    + (size_t)NN * HC)      // N*256
#define OFF_ASRC  (OFF_AGG  + (size_t)NN * HC)      // N*2
#define OFF_ADST  (OFF_ASRC + (size_t)NN * 2)       // N*2
#define OFF_MMAX  (OFF_ADST + (size_t)NN * 2)       // N*2
#define OFF_DEN   (OFF_MMAX + (size_t)NN * 2)       // N*2
#define OFF_Z1    (OFF_DEN  + (size_t)NN * 2)       // N*128
#define OFF_Z2    (OFF_Z1   + (size_t)NN * 128)     // N*64
#define OFF_Z3    (OFF_Z2   + (size_t)NN * 64)      // N*32
#define OFF_ZP    (OFF_Z3   + (size_t)NN * 32)      // N*4 (z padded to 4)
#define OFF_SQ    (OFF_ZP   + (size_t)NN * 4)       // N

__device__ __forceinline__ float leaky(float v, float s) {
    return v >= 0.f ? v : s * v;
}

__device__ __forceinline__ void atomicMaxFloat(float* addr, float val) {
    // bit-ordering trick: int max for non-negative, uint min for negative
    if (val >= 0.f) atomicMax((int*)addr, __float_as_int(val));
    else            atomicMin((unsigned int*)addr, __float_as_uint(val));
}

// CDNA5 async global->LDS copy (ASYNCcnt-tracked; ISA 10.x / 15.18.3)
__device__ __forceinline__ void async_copy_b128(unsigned lds_off, const void* gaddr) {
    asm volatile("global_load_async_to_lds_b128 %0, %1, off"
                 :: "v"(lds_off), "v"(gaddr) : "memory");
}
__device__ __forceinline__ void wait_async0() {
    asm volatile("s_wait_asynccnt 0" ::: "memory");
}

// ---------------- 0: init accumulators ----------------
__global__ void k_init(float* __restrict__ agg, float* __restrict__ mmax,
                       float* __restrict__ denom) {
    int tid = blockIdx.x * 256 + threadIdx.x;
    if (tid < NN * HC) agg[tid] = 0.f;
    if (tid < NN * 2) {
        mmax[tid]  = __int_as_float(0xFF800000);   // -inf
        denom[tid] = 0.f;
    }
}

// ---------------- unified FP32-WMMA GEMM: out = act(A @ W^T (+bias)) --------
// Block = 8 waves covering 128 rows x 16 output features.
// B tile (16 x IN weights) staged once per block via async global->LDS copy.
// PRE: A is read as leaky(A + bpre, 0.01) on the fly (GAT output activation).
// ACT: epilogue leaky(acc + bias, 0.01); otherwise raw store.
template<int IN, bool PRE, bool ACT>
__global__ void k_gemm(const float* __restrict__ A,
                       const float* __restrict__ bpre,
                       const float* __restrict__ W,     // [O][IN] row-major
                       const float* __restrict__ bias,  // [O] (if ACT)
                       float* __restrict__ out, int O) {
    __shared__ float ldsW[16 * IN];                 // 16 weight rows for o-tile
    const int tid  = threadIdx.x;
    const int o0   = blockIdx.y * 16;

    // ---- async-stage the 16 x IN weight tile into LDS (b128 chunks) ----
    constexpr int CH = (16 * IN) / 4;               // float4 chunks
    #pragma unroll
    for (int c = tid; c < CH; c += 256) {
        const int row = c / (IN / 4);
        const int col = c % (IN / 4);
        async_copy_b128((unsigned)(size_t)&ldsW[c * 4],
                        W + (size_t)(o0 + row) * IN + (size_t)col * 4);
    }
    wait_async0();
    __syncthreads();

    const int lane = tid & 31;
    const int w    = tid >> 5;
    const int m    = lane & 15;       // A row within tile / B col within tile
    const int hi   = lane >> 4;       // half-wave selector
    const int i0   = blockIdx.x * 128 + w * 16;

    const float* __restrict__ arow = A + (size_t)(i0 + m) * IN;
    const float* __restrict__ brow = ldsW + m * IN;   // weight row for col m

    v8f c = {};
    for (int kb0 = 0; kb0 < IN; kb0 += 32) {
        if (kb0 + 64 < IN)                            // near-cache prefetch
            __builtin_prefetch(arow + kb0 + 64, 0, 3);
        #pragma unroll
        for (int kb = kb0; kb < kb0 + 32; kb += 4) {
            // A 16x4: lanes0-15 hold K={0,1}, lanes16-31 hold K={2,3}
            v2f a = *(const v2f*)(arow + kb + 2 * hi);
            if (PRE) {
                v2f bp = *(const v2f*)(bpre + kb + 2 * hi);
                a.x = leaky(a.x + bp.x, 0.01f);
                a.y = leaky(a.y + bp.y, 0.01f);
            }
            // B 4x16 fragment from LDS (ds_load_b64)
            v2f b = *(const v2f*)(brow + kb + 2 * hi);
            c = __builtin_amdgcn_wmma_f32_16x16x4_f32(false, a, false, b,
                                                      (short)0, c, false, false);
        }
    }

    // C/D layout: VGPR v -> (M = v + 8*hi, N = lane&15)
    if (ACT) {
        const float bo = bias[o0 + m];
        #pragma unroll
        for (int v = 0; v < 8; ++v)
            out[(size_t)(i0 + v + 8 * hi) * O + o0 + m] = leaky(c[v] + bo, 0.01f);
    } else {
        #pragma unroll
        for (int v = 0; v < 8; ++v)
            out[(size_t)(i0 + v + 8 * hi) * O + o0 + m] = c[v];
    }
}

// ---------------- 2: attention coefficients per node ----------------
__global__ void k_attn(const float* __restrict__ h,
                       const float* __restrict__ att_src,
                       const float* __restrict__ att_dst,
                       float* __restrict__ a_src, float* __restrict__ a_dst) {
    const int lane = threadIdx.x & 31;
    const int w    = threadIdx.x >> 5;
    const int n    = blockIdx.x * 8 + w;
    #pragma unroll
    for (int hh = 0; hh < 2; ++hh) {
        float as = 0.f, ad = 0.f;
        #pragma unroll
        for (int c = lane; c < 128; c += 32) {
            float v = h[(size_t)n * HC + hh * 128 + c];
            as += v * att_src[hh * 128 + c];
            ad += v * att_dst[hh * 128 + c];
        }
        #pragma unroll
        for (int off = 16; off; off >>= 1) {
            as += __shfl_xor(as, off, 32);
            ad += __shfl_xor(ad, off, 32);
        }
        if (lane == 0) { a_src[n * 2 + hh] = as; a_dst[n * 2 + hh] = ad; }
    }
}

__device__ __forceinline__ void edge_sd(const long long* __restrict__ ei,
                                        long long t, long long* s, long long* d) {
    if (t < EE) { *s = ei[t]; *d = ei[EE + t]; }
    else        { *s = t - EE; *d = t - EE; }   // self loop
}

// ---------------- 3: segment max ----------------
__global__ void k_edge_max(const long long* __restrict__ ei,
                           const float* __restrict__ a_src,
                           const float* __restrict__ a_dst,
                           float* __restrict__ mmax) {
    long long t = (long long)blockIdx.x * 256 + threadIdx.x;
    if (t >= TOTE) return;
    long long s, d; edge_sd(ei, t, &s, &d);
    #pragma unroll
    for (int hh = 0; hh < 2; ++hh) {
        float v = leaky(a_src[s * 2 + hh] + a_dst[d * 2 + hh], 0.2f);
        atomicMaxFloat(&mmax[d * 2 + hh], v);
    }
}

// ---------------- 4: segment sum of exp ----------------
__global__ void k_edge_sum(const long long* __restrict__ ei,
                           const float* __restrict__ a_src,
                           const float* __restrict__ a_dst,
                           const float* __restrict__ mmax,
                           float* __restrict__ denom) {
    long long t = (long long)blockIdx.x * 256 + threadIdx.x;
    if (t >= TOTE) return;
    long long s, d; edge_sd(ei, t, &s, &d);
    #pragma unroll
    for (int hh = 0; hh < 2; ++hh) {
        float v = leaky(a_src[s * 2 + hh] + a_dst[d * 2 + hh], 0.2f);
        atomicAdd(&denom[d * 2 + hh], __expf(v - mmax[d * 2 + hh]));
    }
}

// ---------------- 5: weighted aggregation (one wave per edge) ----------------
__global__ void k_edge_aggr(const long long* __restrict__ ei,
                            const float* __restrict__ a_src,
                            const float* __restrict__ a_dst,
                            const float* __restrict__ mmax,
                            const float* __restrict__ denom,
                            const float* __restrict__ h,
                            float* __restrict__ agg) {
    const int lane = threadIdx.x & 31;
    const int w    = threadIdx.x >> 5;
    long long e = (long long)blockIdx.x * 8 + w;
    if (e >= TOTE) return;
    long long s, d; edge_sd(ei, e, &s, &d);

    float al[2];
    #pragma unroll
    for (int hh = 0; hh < 2; ++hh) {
        float v = leaky(a_src[s * 2 + hh] + a_dst[d * 2 + hh], 0.2f);
        al[hh] = __expf(v - mmax[d * 2 + hh]) / (denom[d * 2 + hh] + 1e-16f);
    }

    const float4* __restrict__ hs = (const float4*)(h + (size_t)s * HC);
    float4 v0 = hs[lane];        // channels [4*lane .. 4*lane+3]   -> head 0
    float4 v1 = hs[32 + lane];   // channels [128+4*lane .. +3]     -> head 1
    float* a0 = agg + (size_t)d * HC + 4 * lane;
    float* a1 = a0 + 128;
    atomicAdd(a0 + 0, al[0] * v0.x); atomicAdd(a0 + 1, al[0] * v0.y);
    atomicAdd(a0 + 2, al[0] * v0.z); atomicAdd(a0 + 3, al[0] * v0.w);
    atomicAdd(a1 + 0, al[1] * v1.x); atomicAdd(a1 + 1, al[1] * v1.y);
    atomicAdd(a1 + 2, al[1] * v1.z); atomicAdd(a1 + 3, al[1] * v1.w);
}

// ---------------- 6: final head 32 -> 3 (wave per node) ----------------
__global__ void k_head(const float* __restrict__ z3,   // [N][32], pre-activated
                       const float* __restrict__ W3,   // [3][32]
                       const float* __restrict__ b3,
                       float* __restrict__ zp, float* __restrict__ sq) {
    const int lane = threadIdx.x & 31;
    const int w    = threadIdx.x >> 5;
    const int n    = blockIdx.x * 8 + w;

    const float v = z3[(size_t)n * 32 + lane];
    float zv[3];
    #pragma unroll
    for (int o = 0; o < 3; ++o) {
        float acc = v * W3[o * 32 + lane];
        #pragma unroll
        for (int off = 16; off; off >>= 1) acc += __shfl_xor(acc, off, 32);
        zv[o] = acc + b3[o];
    }
    if (lane == 0) {
        *(float4*)(zp + (size_t)n * 4) = make_float4(zv[0], zv[1], zv[2], 0.f);
        sq[n] = zv[0] * zv[0] + zv[1] * zv[1] + zv[2] * zv[2];
    }
}

// ---------------- 7: cdist via FP32 WMMA (K padded 3->4) ----------------
// block = 8 waves; wave covers 16 rows x 128 cols (8 tiles, A frag reused)
__global__ void k_cdist(const float* __restrict__ zp, const float* __restrict__ sq,
                        float* __restrict__ out) {
    const int lane = threadIdx.x & 31;
    const int w    = threadIdx.x >> 5;
    const int m    = lane & 15;
    const int hi   = lane >> 4;
    const int i0   = blockIdx.y * 128 + w * 16;
    const int j0   = blockIdx.x * 128;

    v2f a = *(const v2f*)(zp + (size_t)(i0 + m) * 4 + 2 * hi);
    float sqi[8];
    #pragma unroll
    for (int v = 0; v < 8; ++v) sqi[v] = sq[i0 + v + 8 * hi];

    #pragma unroll
    for (int jt = 0; jt < 8; ++jt) {
        const int jb = j0 + jt * 16;
        v2f b = *(const v2f*)(zp + (size_t)(jb + m) * 4 + 2 * hi);
        v8f c = {};
        c = __builtin_amdgcn_wmma_f32_16x16x4_f32(false, a, false, b,
                                                  (short)0, c, false, false);
        const float sqj = sq[jb + m];
        #pragma unroll
        for (int v = 0; v < 8; ++v) {
            float d2 = sqi[v] + sqj - 2.f * c[v];
            float dd = d2 > 0.f ? __fsqrt_rn(d2) : 0.f;
            out[(size_t)(i0 + v + 8 * hi) * NN + jb + m] = dd;
        }
    }
}

// ---------------- host launcher ----------------
extern "C" void kernel_launch(void* const* d_in, const int* in_sizes, int n_in,
                              void* d_out, int out_size, void* d_ws, size_t ws_size,
                              hipStream_t stream) {
    const float*     x        = (const float*)d_in[0];
    const long long* ei       = (const long long*)d_in[1];
    const float*     Wg       = (const float*)d_in[2];
    const float*     att_src  = (const float*)d_in[3];
    const float*     att_dst  = (const float*)d_in[4];
    const float*     b_gat    = (const float*)d_in[5];
    const float*     Wa       = (const float*)d_in[6];
    const float*     ba       = (const float*)d_in[7];
    const float*     W1       = (const float*)d_in[8];
    const float*     b1       = (const float*)d_in[9];
    const float*     W2       = (const float*)d_in[10];
    const float*     b2       = (const float*)d_in[11];
    const float*     W3       = (const float*)d_in[12];
    const float*     b3       = (const float*)d_in[13];
    float* out = (float*)d_out;
    float* ws  = (float*)d_ws;

    float* h     = ws + OFF_H;
    float* agg   = ws + OFF_AGG;
    float* a_src = ws + OFF_ASRC;
    float* a_dst = ws + OFF_ADST;
    float* mmax  = ws + OFF_MMAX;
    float* denom = ws + OFF_DEN;
    float* z1    = ws + OFF_Z1;
    float* z2    = ws + OFF_Z2;
    float* z3    = ws + OFF_Z3;
    float* zp    = ws + OFF_ZP;
    float* sq    = ws + OFF_SQ;

    // 0: init accumulators (covers N*256 >= all smaller arrays)
    k_init<<<(NN * HC + 255) / 256, 256, 0, stream>>>(agg, mmax, denom);

    // 1: h = x @ W_gat^T  (WMMA f32, async-staged weight tiles)
    k_gemm<256, false, false><<<dim3(NN / 128, HC / 16), 256, 0, stream>>>(
        x, nullptr, Wg, nullptr, h, HC);

    // 2: attention coefficients
    k_attn<<<NN / 8, 256, 0, stream>>>(h, att_src, att_dst, a_src, a_dst);

    // 3/4: segment softmax stats over dst
    const int eb = (TOTE + 255) / 256;
    k_edge_max<<<eb, 256, 0, stream>>>(ei, a_src, a_dst, mmax);
    k_edge_sum<<<eb, 256, 0, stream>>>(ei, a_src, a_dst, mmax, denom);

    // 5: weighted aggregation (wave per edge)
    k_edge_aggr<<<(TOTE + 7) / 8, 256, 0, stream>>>(ei, a_src, a_dst, mmax, denom, h, agg);

    // 6: MLP layers as WMMA GEMMs
    k_gemm<256, true,  true><<<dim3(NN / 128, 128 / 16), 256, 0, stream>>>(
        agg, b_gat, Wa, ba, z1, 128);
    k_gemm<128, false, true><<<dim3(NN / 128,  64 / 16), 256, 0, stream>>>(
        z1, nullptr, W1, b1, z2, 64);
    k_gemm< 64, false, true><<<dim3(NN / 128,  32 / 16), 256, 0, stream>>>(
        z2, nullptr, W2, b2, z3, 32);

    // 7: head 32 -> 3 + |z|^2
    k_head<<<NN / 8, 256, 0, stream>>>(z3, W3, b3, zp, sq);

    // 8: pairwise distances (WMMA f32, HBM-store bound)
    k_cdist<<<dim3(NN / 128, NN / 128), 256, 0, stream>>>(zp, sq, out);
}